// QRNNLayer_85581518340442
// MI455X (gfx1250) — compile-verified
//
#include <hip/hip_runtime.h>

// ---------------- problem constants ----------------
#define SEQ    4096
#define BATCH  8
#define CIN    256
#define HIDN   256
#define KDIM   512                 // 2*CIN  (concat X[s-1], X[s])
#define MROWS  (SEQ*BATCH)         // 32768
#define NCOLS  512                 // two gates * HIDN
#define COLS   (BATCH*HIDN)        // 2048 scan columns
#define NCHUNK 64
#define CHLEN  (SEQ/NCHUNK)        // 64
#define LDA    516                 // padded LDS row pitch (floats): bank = 4*row mod 64
#define MT     4                   // M-tiles (of 16 rows) per block
#define MBLK   (16*MT)             // 64 rows per block

typedef float v2f __attribute__((ext_vector_type(2)));
typedef float v4f __attribute__((ext_vector_type(4)));
typedef float v8f __attribute__((ext_vector_type(8)));

__device__ __forceinline__ float sigmoidf(float x) { return 1.0f / (1.0f + __expf(-x)); }

// ---------------- kernel 0: weight transpose ----------------
// Wt[n*512 + kk] = W_gate[h, c, k]  with n = gate*256 + h, kk = k*256 + c
__global__ void __launch_bounds__(256) wt_transpose_kernel(
    const float* __restrict__ Wz, const float* __restrict__ Wf, float* __restrict__ Wt)
{
    int t = blockIdx.x * 256 + threadIdx.x;      // 0 .. 512*512-1
    int n  = t >> 9;
    int kk = t & 511;
    int gate = n >> 8;
    int h    = n & 255;
    const float* W = gate ? Wf : Wz;             // (H, C, K) row-major, K=2
    Wt[t] = W[h * 512 + ((kk & 255) << 1) + (kk >> 8)];
}

// ---------------- kernel 1: GEMM (WMMA f32 16x16x4) + activations ----------------
// Each block: 64 M rows (8 seq positions x 8 batches). 8 waves x 4 N-tiles x 4 M-tiles.
// Each B fragment feeds 4 WMMAs -> 4x less L2 traffic on the weight matrix.
// Zact (QuickGELU(z)) -> d_out[s*2048 + b*256 + h]; Fact (sigmoid(f)) -> workspace.
__global__ void __launch_bounds__(256, 1) gemm_act_kernel(
    const float* __restrict__ X,   const float* __restrict__ Wt,
    const float* __restrict__ bz,  const float* __restrict__ bf,
    float* __restrict__ Zact,      float* __restrict__ Fact)
{
    __shared__ float Asm[MBLK * LDA];            // 64*516*4 = 132 KB
    const int tid = threadIdx.x;
    const int m0  = blockIdx.x * MBLK;           // first global row of block

    // Stage A tile: row r = [ X[s-1,b,:] | X[s,b,:] ]  (zeros if s==0)
    for (int i = tid; i < MBLK * 128; i += 256) { // 128 float4 per row
        int r  = i >> 7;
        int c  = (i & 127) << 2;
        int m  = m0 + r;
        v4f val;
        if (c < 256) {
            if (m < BATCH) { val = (v4f){0.f, 0.f, 0.f, 0.f}; }
            else           { val = *(const v4f*)(X + (m - BATCH) * CIN + c); }
        } else {
            val = *(const v4f*)(X + m * CIN + (c - 256));
        }
        *(v4f*)(&Asm[r * LDA + c]) = val;
    }
    __syncthreads();

    const int lane  = tid & 31;
    const int wave  = tid >> 5;
    const int laneM = lane & 15;
    const int koff  = (lane >> 4) << 1;          // lanes 16-31 hold K=k0+2,k0+3

    v8f acc[MT][4] = {};                          // [m-tile][n-tile] 16x16 f32 accumulators

    const float* arow0 = &Asm[laneM * LDA];       // m-tile mt adds mt*16*LDA
    const float* bbase[4];
    #pragma unroll
    for (int t = 0; t < 4; ++t) {
        int n = ((wave << 2) + t) * 16 + laneM;
        bbase[t] = Wt + n * KDIM;
    }

    for (int kk = 0; kk < KDIM; kk += 4) {
        v2f afrag[MT];
        #pragma unroll
        for (int mt = 0; mt < MT; ++mt)
            afrag[mt] = *(const v2f*)(arow0 + mt * (16 * LDA) + kk + koff);

        #pragma unroll
        for (int t = 0; t < 4; ++t) {
            v2f bfrag = *(const v2f*)(bbase[t] + kk + koff);
            #pragma unroll
            for (int mt = 0; mt < MT; ++mt) {
                acc[mt][t] = __builtin_amdgcn_wmma_f32_16x16x4_f32(
                    false, afrag[mt], false, bfrag, (short)0, acc[mt][t], false, false);
            }
        }
    }

    // Epilogue: bias + activation, straight from accumulators.
    const int hi8 = (lane >> 4) << 3;             // lanes>=16 hold M = j+8
    #pragma unroll
    for (int t = 0; t < 4; ++t) {
        int n    = ((wave << 2) + t) * 16 + laneM;
        int gate = n >> 8;
        int h    = n & 255;
        float bias = gate ? bf[h] : bz[h];
        #pragma unroll
        for (int mt = 0; mt < MT; ++mt) {
            int mbase = m0 + mt * 16 + hi8;
            #pragma unroll
            for (int j = 0; j < 8; ++j) {
                float v = acc[mt][t][j] + bias;
                int   m = mbase + j;
                if (gate == 0) {                  // z-gate: QuickGELU
                    Zact[m * HIDN + h] = v * sigmoidf(1.702f * v);
                } else {                          // f-gate: sigmoid
                    Fact[m * HIDN + h] = sigmoidf(v);
                }
            }
        }
    }
}

// ---------------- kernel 2: per-chunk affine composition ----------------
// h' = f*z + (1-f)*h  ==  h' = a*h + b, a = 1-f, b = f*z.  Compose over CHLEN steps.
__global__ void __launch_bounds__(256) chunk_compose_kernel(
    const float* __restrict__ Zact, const float* __restrict__ Fact,
    float* __restrict__ chA, float* __restrict__ chB)
{
    int g     = blockIdx.x * 256 + threadIdx.x;  // 0 .. NCHUNK*COLS-1
    int chunk = g >> 11;
    int col   = g & (COLS - 1);
    float A = 1.f, Bv = 0.f;
    int idx = chunk * CHLEN * COLS + col;
    for (int i = 0; i < CHLEN; ++i, idx += COLS) {
        float f = Fact[idx];
        float z = Zact[idx];
        float a = 1.f - f;
        Bv = a * Bv + f * z;
        A *= a;
    }
    chA[chunk * COLS + col] = A;
    chB[chunk * COLS + col] = Bv;
}

// ---------------- kernel 3: sequential cross-chunk prefix (tiny) ----------------
__global__ void __launch_bounds__(256) chunk_prefix_kernel(
    const float* __restrict__ hidden, const float* __restrict__ chA,
    const float* __restrict__ chB, float* __restrict__ chH0, float* __restrict__ hlast)
{
    int col = blockIdx.x * 256 + threadIdx.x;    // 0 .. COLS-1
    float h = hidden[col];
    for (int c = 0; c < NCHUNK; ++c) {
        chH0[c * COLS + col] = h;
        h = chA[c * COLS + col] * h + chB[c * COLS + col];
    }
    hlast[col] = h;                               // d_out tail: (1,B,H)
}

// ---------------- kernel 4: apply local scan, write H in-place over Zact ----------------
__global__ void __launch_bounds__(256) chunk_apply_kernel(
    const float* __restrict__ Fact, const float* __restrict__ chH0,
    float* __restrict__ ZH)                       // d_out: reads z, overwrites with h
{
    int g     = blockIdx.x * 256 + threadIdx.x;
    int chunk = g >> 11;
    int col   = g & (COLS - 1);
    float h = chH0[chunk * COLS + col];
    int idx = chunk * CHLEN * COLS + col;
    for (int i = 0; i < CHLEN; ++i, idx += COLS) {
        float f = Fact[idx];
        float z = ZH[idx];
        h = f * z + (1.f - f) * h;
        ZH[idx] = h;
    }
}

// ---------------- launcher ----------------
extern "C" void kernel_launch(void* const* d_in, const int* in_sizes, int n_in,
                              void* d_out, int out_size, void* d_ws, size_t ws_size,
                              hipStream_t stream)
{
    const float* X      = (const float*)d_in[0];
    const float* hidden = (const float*)d_in[1];
    const float* Wz     = (const float*)d_in[2];
    const float* bz     = (const float*)d_in[3];
    const float* Wf     = (const float*)d_in[4];
    const float* bf     = (const float*)d_in[5];

    float* out   = (float*)d_out;                 // [S*B*H] H  +  [B*H] h_last
    float* hlast = out + (size_t)MROWS * HIDN;

    // workspace layout (floats)
    float* ws   = (float*)d_ws;
    float* Wt   = ws;                             // 512*512          = 262144
    float* Fact = Wt   + (size_t)NCOLS * KDIM;    // S*B*H            = 8388608
    float* chA  = Fact + (size_t)MROWS * HIDN;    // NCHUNK*COLS      = 131072
    float* chB  = chA  + (size_t)NCHUNK * COLS;
    float* chH0 = chB  + (size_t)NCHUNK * COLS;

    wt_transpose_kernel<<<(NCOLS * KDIM) / 256, 256, 0, stream>>>(Wz, Wf, Wt);

    gemm_act_kernel<<<MROWS / MBLK, 256, 0, stream>>>(X, Wt, bz, bf, /*Zact=*/out, Fact);

    chunk_compose_kernel<<<(NCHUNK * COLS) / 256, 256, 0, stream>>>(out, Fact, chA, chB);

    chunk_prefix_kernel<<<COLS / 256, 256, 0, stream>>>(hidden, chA, chB, chH0, hlast);

    chunk_apply_kernel<<<(NCHUNK * COLS) / 256, 256, 0, stream>>>(Fact, chH0, out);
}